// DirectionalConv_53017076301933
// MI455X (gfx1250) — compile-verified
//
#include <hip/hip_runtime.h>

#define BLK   256
#define CHUNK 256
#define FEAT  32

#if __has_builtin(__builtin_amdgcn_global_load_async_to_lds_b32) && \
    __has_builtin(__builtin_amdgcn_s_wait_asynccnt)
#define USE_ASYNC 1
#else
#define USE_ASYNC 0
#endif

#if USE_ASYNC
typedef __attribute__((address_space(1))) int gint_t;   // global (addrspace 1)
typedef __attribute__((address_space(3))) int lint_t;   // LDS    (addrspace 3)
__device__ __forceinline__ void async_copy_b32(void* lds_dst, const void* g_src) {
  // global_load_async_to_lds_b32 : per-lane 4B, tracked by ASYNCcnt
  __builtin_amdgcn_global_load_async_to_lds_b32((gint_t*)g_src, (lint_t*)lds_dst, 0, 0);
}
#endif

// One edge per wave32: lane == feature index (F == 32 == wave width).
// Edge metadata (row/col/w) is staged into LDS in CHUNK-sized tiles with
// async global->LDS copies, double-buffered across the persistent chunk loop.
// Each lane then owns one edge's metadata; v_readlane broadcasts it to SGPRs
// so x/out addresses become SALU-computed saddr bases.
__global__ __launch_bounds__(BLK) void gconv_scatter(
    const float* __restrict__ x,
    const int*   __restrict__ row,
    const int*   __restrict__ col,
    const float* __restrict__ wgt,
    float*       __restrict__ out,
    int E, int nChunks) {
  __shared__ int   rS[2][CHUNK];
  __shared__ int   cS[2][CHUNK];
  __shared__ float wS[2][CHUNK];

  const int t      = threadIdx.x;
  const int lane   = t & 31;
  const int waveId = t >> 5;
  const int stride = gridDim.x;

  int c = blockIdx.x;
  if (c >= nChunks) return;

  // Prologue: stage first chunk into buffer 0 (clamped index keeps the async
  // instruction count per wave exact even on the tail chunk).
  {
    int src = c * CHUNK + t;
    if (src >= E) src = E - 1;
#if USE_ASYNC
    async_copy_b32(&rS[0][t], &row[src]);
    async_copy_b32(&cS[0][t], &col[src]);
    async_copy_b32(&wS[0][t], &wgt[src]);
#else
    rS[0][t] = row[src];
    cS[0][t] = col[src];
    wS[0][t] = wgt[src];
#endif
  }

  int cur = 0;
  for (; c < nChunks; c += stride) {
    const int  cn      = c + stride;
    const bool hasNext = (cn < nChunks);   // block-uniform
    if (hasNext) {
      int src = cn * CHUNK + t;
      if (src >= E) src = E - 1;
#if USE_ASYNC
      async_copy_b32(&rS[cur ^ 1][t], &row[src]);
      async_copy_b32(&cS[cur ^ 1][t], &col[src]);
      async_copy_b32(&wS[cur ^ 1][t], &wgt[src]);
#else
      rS[cur ^ 1][t] = row[src];
      cS[cur ^ 1][t] = col[src];
      wS[cur ^ 1][t] = wgt[src];
#endif
    }
#if USE_ASYNC
    // 3 async instructions per stage per wave; leave the 3 just-issued
    // (next chunk) outstanding, guarantee current chunk's 3 completed.
    if (hasNext) __builtin_amdgcn_s_wait_asynccnt(3);
    else         __builtin_amdgcn_s_wait_asynccnt(0);
#endif
    __syncthreads();

    const int base = c * CHUNK + waveId * 32;  // this wave's 32 edges
    const int lidx = waveId * 32;

    // Lane j holds edge (base + j)'s metadata: 3 per-lane DS loads total.
    const int   myR = rS[cur][lidx + lane];
    const int   myC = cS[cur][lidx + lane];
    const int   myW = __float_as_int(wS[cur][lidx + lane]);

#pragma unroll
    for (int j = 0; j < 32; ++j) {
      const int e = base + j;                  // wave-uniform guard
      if (e < E) {
        const int   r  = __builtin_amdgcn_readlane(myR, j);  // -> SGPR
        const int   cc = __builtin_amdgcn_readlane(myC, j);  // -> SGPR
        const float wv = __int_as_float(__builtin_amdgcn_readlane(myW, j));
        const float xv = x[(size_t)cc * FEAT + lane];        // 1 cacheline / wave
        unsafeAtomicAdd(&out[(size_t)r * FEAT + lane], xv * wv); // global_atomic_add_f32
      }
    }
    __syncthreads();   // buffer reusable for the stage issued next iteration
    cur ^= 1;
  }
}

__global__ __launch_bounds__(BLK) void gconv_zero(float4* out4, int n4) {
  int i = blockIdx.x * BLK + threadIdx.x;
  if (i < n4) out4[i] = make_float4(0.f, 0.f, 0.f, 0.f);
}

__global__ __launch_bounds__(BLK) void gconv_scale(
    float4* __restrict__ out4, const float* __restrict__ deg_inv, int n4) {
  int i = blockIdx.x * BLK + threadIdx.x;
  if (i < n4) {
    const float d = deg_inv[i / (FEAT / 4)];
    float4 v = out4[i];
    v.x *= d; v.y *= d; v.z *= d; v.w *= d;
    out4[i] = v;
  }
}

extern "C" void kernel_launch(void* const* d_in, const int* in_sizes, int n_in,
                              void* d_out, int out_size, void* d_ws, size_t ws_size,
                              hipStream_t stream) {
  const float* x       = (const float*)d_in[0];
  const int*   eidx    = (const int*)  d_in[1];
  const float* wgt     = (const float*)d_in[2];
  const float* deg_inv = (const float*)d_in[3];
  float*       out     = (float*)d_out;

  const int E = in_sizes[2];          // number of edges
  const int* row = eidx;              // edge_index[0]
  const int* col = eidx + E;          // edge_index[1]

  const int n4 = out_size / 4;        // N*FEAT / 4
  const int zb = (n4 + BLK - 1) / BLK;
  gconv_zero<<<zb, BLK, 0, stream>>>((float4*)out, n4);

  const int nChunks = (E + CHUNK - 1) / CHUNK;
  int blocks = nChunks < 1250 ? nChunks : 1250;   // persistent, ~5 chunks/block
  gconv_scatter<<<blocks, BLK, 0, stream>>>(x, row, col, wgt, out, E, nChunks);

  gconv_scale<<<zb, BLK, 0, stream>>>((float4*)out, deg_inv, n4);
}